// GeneralizedCrossEntropy_21517786153522
// MI455X (gfx1250) — compile-verified
//
#include <hip/hip_runtime.h>
#include <hip/hip_bf16.h>
#include <stdint.h>

#ifndef __has_builtin
#define __has_builtin(x) 0
#endif

#if __has_builtin(__builtin_amdgcn_global_load_async_to_lds_b128)
#define USE_ASYNC_LDS 1
#else
#define USE_ASYNC_LDS 0
#endif

#if __has_builtin(__builtin_amdgcn_s_wait_asynccnt)
#define WAIT_ASYNCCNT(n) do { __builtin_amdgcn_s_wait_asynccnt(n); asm volatile("" ::: "memory"); } while (0)
#else
#define WAIT_ASYNCCNT(n) asm volatile("s_wait_asynccnt %0" :: "i"(n) : "memory")
#endif

// The b128 async builtin takes typed 16-byte-vector pointers:
//   (v4i addrspace(1)* gaddr, v4i addrspace(3)* laddr, imm offset, imm cpol)
typedef int v4i __attribute__((vector_size(16)));
#define AS1_V4I(p) ((__attribute__((address_space(1))) v4i*)(uintptr_t)(p))
#define AS3_V4I(p) ((__attribute__((address_space(3))) v4i*)(uintptr_t)(p))

namespace {
constexpr int   kB      = 512;
constexpr int   kC      = 4;
constexpr int   kS      = 16384;
constexpr int   kPAD    = 3;
constexpr float kQ      = 0.7f;
constexpr float kInvQ   = 1.0f / 0.7f;
constexpr float kL2E    = 1.4426950408889634f;   // log2(e)
constexpr float kLogClip = -23.253496664211536f; // log2(1e-7)

constexpr int TILE    = 1024;          // positions per pipeline tile
constexpr int TILES   = kS / TILE;     // 16
constexpr int THREADS = 256;           // 8 wave32
}

// Hardware transcendentals (v_exp_f32 = exp2, v_log_f32 = log2).
__device__ __forceinline__ float fast_exp2(float x) { return __builtin_amdgcn_exp2f(x); }
__device__ __forceinline__ float fast_log2(float x) { return __builtin_amdgcn_logf(x); }

// One position: softmax over 3 channels in log2 domain, GCE term.
__device__ __forceinline__ void gce1(int lab, float c0, float c1, float c2, int s,
                                     float& sum, int& minPad) {
  if (lab == kPAD) { minPad = min(minPad, s); return; }
  float m  = fmaxf(c0, fmaxf(c1, c2));
  float e0 = fast_exp2((c0 - m) * kL2E);
  float e1 = fast_exp2((c1 - m) * kL2E);
  float e2 = fast_exp2((c2 - m) * kL2E);
  float dl = fast_log2(e0 + e1 + e2);             // log2(denominator)
  float xl = (lab == 0) ? c0 : ((lab == 1) ? c1 : c2);
  float lp = (xl - m) * kL2E - dl;                // log2(p_true), <= 0
  lp = fmaxf(lp, kLogClip);                       // clip p >= 1e-7
  float p07 = fast_exp2(kQ * lp);                 // p^0.7
  sum += (1.0f - p07) * kInvQ;
}

__global__ void __launch_bounds__(THREADS)
gce_row_kernel(const float* __restrict__ pred,
               const int*   __restrict__ labels,
               float*       __restrict__ per_sample) {
  const int b   = blockIdx.x;
  const int tid = threadIdx.x;
  const int wave = tid >> 5;
  const int lane = tid & 31;
  const int e = wave * 128 + lane * 4;            // this lane's slot inside a tile

  const size_t rowL = (size_t)b * kS;
  const size_t rowP = (size_t)b * kC * kS;

  __shared__ float s_sum[THREADS];
  __shared__ int   s_min[THREADS];

  float sum    = 0.0f;
  int   minPad = 0x7fffffff;

#if USE_ASYNC_LDS
  __shared__ int   s_lab[2][TILE];                //  8 KB
  __shared__ float s_prd[2][3][TILE];             // 24 KB

  // Prologue: DMA tile 0 into buffer 0 (4 async b128 per lane).
  {
    const int s0 = e;
    __builtin_amdgcn_global_load_async_to_lds_b128(AS1_V4I(labels + rowL + s0),
                                                   AS3_V4I(&s_lab[0][e]), 0, 0);
#pragma unroll
    for (int c = 0; c < 3; ++c)
      __builtin_amdgcn_global_load_async_to_lds_b128(AS1_V4I(pred + rowP + (size_t)c * kS + s0),
                                                     AS3_V4I(&s_prd[0][c][e]), 0, 0);
  }

  for (int t = 0; t < TILES; ++t) {
    const int cur = t & 1;
    if (t + 1 < TILES) {
      const int nxt = (t + 1) & 1;
      const int s0  = (t + 1) * TILE + e;
      __builtin_amdgcn_global_load_async_to_lds_b128(AS1_V4I(labels + rowL + s0),
                                                     AS3_V4I(&s_lab[nxt][e]), 0, 0);
#pragma unroll
      for (int c = 0; c < 3; ++c)
        __builtin_amdgcn_global_load_async_to_lds_b128(AS1_V4I(pred + rowP + (size_t)c * kS + s0),
                                                       AS3_V4I(&s_prd[nxt][c][e]), 0, 0);
      WAIT_ASYNCCNT(4);   // tile t landed; tile t+1 still in flight
    } else {
      WAIT_ASYNCCNT(0);   // drain for the last tile
    }

    // Each lane consumes exactly the 16B regions it DMA'd -> no block barrier needed.
    int4   lb = *(const int4*)  (&s_lab[cur][e]);
    float4 x0 = *(const float4*)(&s_prd[cur][0][e]);
    float4 x1 = *(const float4*)(&s_prd[cur][1][e]);
    float4 x2 = *(const float4*)(&s_prd[cur][2][e]);

    const int sb = t * TILE + e;
    gce1(lb.x, x0.x, x1.x, x2.x, sb + 0, sum, minPad);
    gce1(lb.y, x0.y, x1.y, x2.y, sb + 1, sum, minPad);
    gce1(lb.z, x0.z, x1.z, x2.z, sb + 2, sum, minPad);
    gce1(lb.w, x0.w, x1.w, x2.w, sb + 3, sum, minPad);
  }
#else
  // Fallback: direct vectorized global loads; skip pred loads for all-PAD groups (tail).
  for (int t = 0; t < TILES; ++t) {
    const int sb = t * TILE + e;
    int4 lb = *(const int4*)(labels + rowL + sb);
    if (lb.x == kPAD && lb.y == kPAD && lb.z == kPAD && lb.w == kPAD) {
      minPad = min(minPad, sb);
      continue;
    }
    float4 x0 = *(const float4*)(pred + rowP + (size_t)0 * kS + sb);
    float4 x1 = *(const float4*)(pred + rowP + (size_t)1 * kS + sb);
    float4 x2 = *(const float4*)(pred + rowP + (size_t)2 * kS + sb);
    gce1(lb.x, x0.x, x1.x, x2.x, sb + 0, sum, minPad);
    gce1(lb.y, x0.y, x1.y, x2.y, sb + 1, sum, minPad);
    gce1(lb.z, x0.z, x1.z, x2.z, sb + 2, sum, minPad);
    gce1(lb.w, x0.w, x1.w, x2.w, sb + 3, sum, minPad);
  }
#endif

  // Deterministic block reduction: sum and first-pad index.
  s_sum[tid] = sum;
  s_min[tid] = minPad;
  __syncthreads();
  for (int off = THREADS / 2; off > 0; off >>= 1) {
    if (tid < off) {
      s_sum[tid] += s_sum[tid + off];
      s_min[tid]  = min(s_min[tid], s_min[tid + off]);
    }
    __syncthreads();
  }
  if (tid == 0) {
    const int j = (s_min[0] == 0x7fffffff) ? (kS - 1) : s_min[0];
    per_sample[b] = s_sum[0] / (float)j;
  }
}

// Fixed-order final reduction over 512 per-sample values -> scalar mean.
__global__ void __launch_bounds__(256)
gce_final_kernel(const float* __restrict__ per_sample, float* __restrict__ out) {
  __shared__ float s[256];
  const int tid = threadIdx.x;
  s[tid] = per_sample[tid] + per_sample[tid + 256];
  __syncthreads();
  for (int off = 128; off > 0; off >>= 1) {
    if (tid < off) s[tid] += s[tid + off];
    __syncthreads();
  }
  if (tid == 0) out[0] = s[0] * (1.0f / (float)kB);
}

extern "C" void kernel_launch(void* const* d_in, const int* in_sizes, int n_in,
                              void* d_out, int out_size, void* d_ws, size_t ws_size,
                              hipStream_t stream) {
  (void)in_sizes; (void)n_in; (void)out_size; (void)ws_size;
  const float* pred   = (const float*)d_in[0];
  const int*   labels = (const int*)d_in[1];
  float* per_sample   = (float*)d_ws;            // 512 floats of scratch
  float* out          = (float*)d_out;

  gce_row_kernel<<<dim3(kB), dim3(THREADS), 0, stream>>>(pred, labels, per_sample);
  gce_final_kernel<<<dim3(1), dim3(256), 0, stream>>>(per_sample, out);
}